// ECS_explicit_pred_3D_49495203119264
// MI455X (gfx1250) — compile-verified
//
#include <hip/hip_runtime.h>

// ECS explicit-pred 3D: fused single-pass depth-stencil streaming kernel.
// B=8, D=107, H=W=256. 224 MB in / 2 MB out -> HBM-bound (~10us floor at
// 23.3 TB/s). One thread per (b,h,w) column; depth slabs are staged into LDS
// with CDNA5 async global->LDS copies (ASYNCcnt), double buffered. Each lane
// only reads back LDS words its own async loads wrote, so no barriers are
// needed -- just s_wait_asynccnt. Streamed data is single-use -> TH_LOAD_NT;
// output is write-once -> non-temporal store.

constexpr int Bb   = 8;
constexpr int Dd   = 107;
constexpr int Hh   = 256;
constexpr int Ww   = 256;
constexpr int HW   = Hh * Ww;           // 65536
constexpr int SLAB = 16;                // depth slices per stage (16 KB LDS)
// stages: 16,16,16,16,16,16,11  (7 stages cover 107 slices)

__device__ __forceinline__ float sgnf(float d) {
  float p = (d > 0.0f) ? 1.0f : 0.0f;
  float n = (d < 0.0f) ? 1.0f : 0.0f;
  return p - n;   // exact {-1,0,1}
}

// CDNA5 async copy: each lane copies 4 bytes from global address g to LDS byte offset lds_byte_off, ASYNCcnt++.
// Non-temporal: the streamed tensor is read exactly once (no reuse to cache).
__device__ __forceinline__ void async_ld_b32(unsigned lds_byte_off,
                                             const float* g) {
  asm volatile("global_load_async_to_lds_b32 %0, %1, off th:TH_LOAD_NT"
               :
               : "v"(lds_byte_off), "v"((unsigned long long)g)
               : "memory");
}

// Wait until ASYNCcnt <= n (async loads complete in order).
#define WAIT_ASYNCCNT(n) asm volatile("s_wait_asynccnt " #n ::: "memory")

__global__ __launch_bounds__(256) void ecs_kernel(
    const float* __restrict__ ssp,
    const float* __restrict__ depth_array,
    float* __restrict__ out) {
  const int tid = threadIdx.x;
  const unsigned col = blockIdx.x * 256u + tid;     // 0 .. B*H*W-1
  const unsigned b   = col >> 16;                   // / HW
  const unsigned rem = col & 65535u;                // % HW
  const float* base = ssp + (size_t)b * (size_t)Dd * (size_t)HW + rem;

  __shared__ float buf[2][SLAB][256];   // 32 KB double buffer
  __shared__ float dep[Dd];
  if (tid < Dd) dep[tid] = (tid == 0) ? 0.0f : depth_array[tid];
  __syncthreads();

  // Issue one stage of async copies: slice d -> buf[(d/SLAB)&1][d%SLAB][tid]
  auto issue = [&](int s) {
    const int d0 = s * SLAB;
    const int n  = (d0 + SLAB <= Dd) ? SLAB : (Dd - d0);
    const int p  = s & 1;
#pragma unroll
    for (int j = 0; j < SLAB; ++j) {
      if (j < n) {
        async_ld_b32((unsigned)(size_t)&buf[p][j][tid],
                     base + (size_t)(d0 + j) * (size_t)HW);
      }
    }
  };
  auto X = [&](int d) -> float { return buf[(d / SLAB) & 1][d % SLAB][tid]; };

  // Sliding sign window: sm1=s[k-1], s0=s[k], s1=s[k+1], s2=s[k+2]
  float sm1 = 0.0f, s0 = 0.0f, s1 = 0.0f, s2 = 0.0f;
  float csum = 0.0f, best = 0.0f, xprev = 0.0f;

  // One output-depth step k (0..106). Out-of-range signs are zero-filled;
  // padding of the pattern masks (pr=1.0) handled by the k-range flags, which
  // are compile-time constants after full unroll.
  auto step = [&](int k) {
    float Sm1 = (k >= 1)   ? sm1 : 0.0f;   // sp zero-pad at front
    float Sk  = (k <= 105) ? s0  : 0.0f;   // sp zero-pad at back
    float sd  = Sk - Sm1;                  // sign_diff[k]
    // element_match for patterns [1,0,1] and [1,-1,0,0] at j=k-1
    float t0  = sm1 - 1.0f;
    float q   = t0 * t0;
    float t1  = s1 - 1.0f;
    float pr1 = q + s0 * s0 + t1 * t1;
    float t2  = s0 + 1.0f;
    float pr2 = q + t2 * t2 + s1 * s1 + s2 * s2;
    if (!(k >= 1 && k <= 104)) pr1 = 1.0f;   // pad(1, n-1) with 1.0
    if (!(k >= 1 && k <= 103)) pr2 = 1.0f;
    // mask_disc = 1 - relu(pr+1)*relu(1-pr); pr >= 0 so relu(pr+1) == pr+1
    float m1 = 1.0f - (pr1 + 1.0f) * fmaxf(1.0f - pr1, 0.0f);
    float m2 = 1.0f - (pr2 + 1.0f) * fmaxf(1.0f - pr2, 0.0f);
    // tanh(10*relu(-sd)) for sd in {-2,-1,0,1,2}: rounds to 1.0f iff sd < 0
    float schg = (sd < 0.0f) ? (m1 * m2) : 0.0f;
    csum += schg;
    float mask = fmaxf(2.0f - csum, 0.0f);
    float ecs  = schg * mask * dep[k];       // dep[0] pre-zeroed
    best = fmaxf(best, ecs);
    sm1 = s0; s0 = s1; s1 = s2;              // shift window
  };

  // Consume one staged slab: step k=d-4 consumes lookahead X(d) -> s[d-1]
  auto consume = [&](int d0, int n) {
#pragma unroll
    for (int j = 0; j < SLAB; ++j) {
      if (j < n) {
        const int d = d0 + j;
        step(d - 4);
        float xn = X(d);
        s2 = sgnf(xn - xprev);
        xprev = xn;
      }
    }
  };

  // ---- software pipeline over 7 stages (double-buffered async copies) ----
  issue(0);
  issue(1);
  WAIT_ASYNCCNT(16);          // stage 0 resident (stage 1 still in flight)
  {                           // prologue: prime window from X(0..3)
    float x0 = X(0), x1 = X(1), x2 = X(2), x3 = X(3);
    s0 = sgnf(x1 - x0); s1 = sgnf(x2 - x1); s2 = sgnf(x3 - x2);
    xprev = x3;
#pragma unroll
    for (int d = 4; d < 16; ++d) {
      step(d - 4);
      float xn = X(d);
      s2 = sgnf(xn - xprev);
      xprev = xn;
    }
  }
  issue(2); WAIT_ASYNCCNT(16); consume(16, 16);
  issue(3); WAIT_ASYNCCNT(16); consume(32, 16);
  issue(4); WAIT_ASYNCCNT(16); consume(48, 16);
  issue(5); WAIT_ASYNCCNT(16); consume(64, 16);
  issue(6); WAIT_ASYNCCNT(11); consume(80, 16);
  WAIT_ASYNCCNT(0);            consume(96, 11);

  // epilogue: k = 103..106 (no more data; s zero-fill past s[105])
  step(103); s2 = 0.0f;
  step(104); s2 = 0.0f;
  step(105); s2 = 0.0f;
  step(106);

  // write-once output: non-temporal store
  __builtin_nontemporal_store(best / 670.25141631f, &out[col]);
}

extern "C" void kernel_launch(void* const* d_in, const int* in_sizes, int n_in,
                              void* d_out, int out_size, void* d_ws,
                              size_t ws_size, hipStream_t stream) {
  const float* ssp   = (const float*)d_in[0];
  const float* depth = (const float*)d_in[1];
  float* out = (float*)d_out;
  const int n_cols = Bb * Hh * Ww;            // 524288
  ecs_kernel<<<dim3(n_cols / 256), dim3(256), 0, stream>>>(ssp, depth, out);
}